// SAGEConv_67053029425276
// MI455X (gfx1250) — compile-verified
//
#include <hip/hip_runtime.h>
#include <hip/hip_bf16.h>

typedef __attribute__((ext_vector_type(2))) float v2f;
typedef __attribute__((ext_vector_type(8))) float v8f;

#define D 128  // D_IN == D_OUT == 128

// ---------------------------------------------------------------------------
// Kernel 1: zero the aggregation scratch (float4-wide grid-stride).
// ---------------------------------------------------------------------------
__global__ __launch_bounds__(256) void sage_zero_kernel(float4* __restrict__ p, long n4) {
    long i = (long)blockIdx.x * blockDim.x + threadIdx.x;
    long stride = (long)gridDim.x * blockDim.x;
    float4 z = make_float4(0.f, 0.f, 0.f, 0.f);
    for (; i < n4; i += stride) p[i] = z;
}

// ---------------------------------------------------------------------------
// Kernel 2: edge scatter. One wave32 per edge; lane l handles 4 contiguous
// floats (float4 load of x[src], 4x global_atomic_add_f32 into agg[dst]).
// ---------------------------------------------------------------------------
__global__ __launch_bounds__(256) void sage_scatter_kernel(
    const float* __restrict__ x, const int* __restrict__ edge_src,
    const int* __restrict__ edge_dst, const float* __restrict__ edge_weight,
    float* __restrict__ agg, int E)
{
    int gid  = blockIdx.x * blockDim.x + threadIdx.x;
    int e    = gid >> 5;
    int lane = gid & 31;
    if (e >= E) return;
    int   src = edge_src[e];
    int   dst = edge_dst[e];
    float w   = edge_weight[e];
    float4 v = ((const float4*)(x + (size_t)src * D))[lane];   // 512B coalesced per wave
    float* ap = agg + (size_t)dst * D + lane * 4;
    atomicAdd(ap + 0, v.x * w);
    atomicAdd(ap + 1, v.y * w);
    atomicAdd(ap + 2, v.z * w);
    atomicAdd(ap + 3, v.w * w);
}

// ---------------------------------------------------------------------------
// Kernel 3: fused dual GEMM + bias + L1 normalize.
// One wave computes a 16x128 output row block:
//   C = agg[16,128] @ W_l + x[16,128] @ W_r   via v_wmma_f32_16x16x4_f32
// A-frag layout (f32 16x4): lanes 0-15 -> row=lane, K={k0,k0+1};
//                           lanes 16-31 -> row=lane-16, K={k0+2,k0+3}.
// B-frag layout (f32 4x16): lanes 0-15 -> col=lane,   rows K={k0,k0+1};
//                           lanes 16-31 -> col=lane-16, rows K={k0+2,k0+3}.
// C/D layout: VGPR j holds row j (lanes 0-15) and row j+8 (lanes 16-31).
// ---------------------------------------------------------------------------
__global__ __launch_bounds__(256) void sage_gemm_l1_kernel(
    const float* __restrict__ agg, const float* __restrict__ x,
    const float* __restrict__ Wl,  const float* __restrict__ bl,
    const float* __restrict__ Wr,  const float* __restrict__ br,
    float* __restrict__ out, int nrows)
{
    const int wid  = blockIdx.x * (blockDim.x >> 5) + (threadIdx.x >> 5);
    const int row0 = wid * 16;
    if (row0 >= nrows) return;            // wave-uniform: EXEC all-1s for live waves

    const int lane = threadIdx.x & 31;
    const int half = lane >> 4;           // 0: lanes 0-15, 1: lanes 16-31
    const int ln   = lane & 15;

    v8f c[8] = {};                        // 8 column tiles of C (16x16 each)

    // ---- pass 1: agg @ W_l ----
    {
        const float* aBase = agg + (size_t)(row0 + ln) * D + half * 2;
        for (int k0 = 0; k0 < D; k0 += 4) {
            v2f a = *(const v2f*)(aBase + k0);
            const float* wp = Wl + (size_t)(k0 + half * 2) * D + ln;
            #pragma unroll
            for (int t = 0; t < 8; ++t) {
                v2f b;
                b.x = wp[16 * t];
                b.y = wp[16 * t + D];
                c[t] = __builtin_amdgcn_wmma_f32_16x16x4_f32(
                    false, a, false, b, (short)0, c[t], false, false);
            }
        }
    }
    // ---- pass 2: += x @ W_r ----
    {
        const float* xBase = x + (size_t)(row0 + ln) * D + half * 2;
        for (int k0 = 0; k0 < D; k0 += 4) {
            v2f a = *(const v2f*)(xBase + k0);
            const float* wp = Wr + (size_t)(k0 + half * 2) * D + ln;
            #pragma unroll
            for (int t = 0; t < 8; ++t) {
                v2f b;
                b.x = wp[16 * t];
                b.y = wp[16 * t + D];
                c[t] = __builtin_amdgcn_wmma_f32_16x16x4_f32(
                    false, a, false, b, (short)0, c[t], false, false);
            }
        }
    }

    // ---- bias + per-row abs partial sums ----
    float s[8];
    #pragma unroll
    for (int j = 0; j < 8; ++j) s[j] = 0.f;
    #pragma unroll
    for (int t = 0; t < 8; ++t) {
        float bb = bl[16 * t + ln] + br[16 * t + ln];
        #pragma unroll
        for (int j = 0; j < 8; ++j) {
            float v = c[t][j] + bb;
            c[t][j] = v;
            s[j] += fabsf(v);
        }
    }

    // ---- L1 reduce within each 16-lane half, scale, store ----
    #pragma unroll
    for (int j = 0; j < 8; ++j) {
        float r = s[j];
        r += __shfl_xor(r, 1, 32);
        r += __shfl_xor(r, 2, 32);
        r += __shfl_xor(r, 4, 32);
        r += __shfl_xor(r, 8, 32);        // masks <16: halves stay independent
        float inv = 1.0f / fmaxf(r, 1e-12f);
        const int row = row0 + j + half * 8;
        float* op = out + (size_t)row * D + ln;
        #pragma unroll
        for (int t = 0; t < 8; ++t) op[16 * t] = c[t][j] * inv;
    }
}

// ---------------------------------------------------------------------------
extern "C" void kernel_launch(void* const* d_in, const int* in_sizes, int n_in,
                              void* d_out, int out_size, void* d_ws, size_t ws_size,
                              hipStream_t stream) {
    const float* x   = (const float*)d_in[0];
    const int*   es  = (const int*)  d_in[1];
    const int*   ed  = (const int*)  d_in[2];
    const float* ew  = (const float*)d_in[3];
    const float* Wl  = (const float*)d_in[4];
    const float* bl  = (const float*)d_in[5];
    const float* Wr  = (const float*)d_in[6];
    const float* br  = (const float*)d_in[7];
    float*       out = (float*)d_out;

    const int N = in_sizes[0] / D;   // 100000
    const int E = in_sizes[1];       // 1600000

    float* agg = (float*)d_ws;       // N*D floats = 51.2 MB scratch

    // 1) zero agg
    {
        long n4 = (long)N * (D / 4);
        int blocks = (int)((n4 + 255) / 256);
        if (blocks > 32768) blocks = 32768;
        sage_zero_kernel<<<blocks, 256, 0, stream>>>((float4*)agg, n4);
    }
    // 2) edge scatter-add (one wave32 per edge)
    {
        long threads = (long)E * 32;
        int blocks = (int)((threads + 255) / 256);
        sage_scatter_kernel<<<blocks, 256, 0, stream>>>(x, es, ed, ew, agg, E);
    }
    // 3) fused GEMM + bias + L1 normalize (one wave per 16 rows)
    {
        int waves  = (N + 15) / 16;
        int blocks = (waves + 7) / 8;    // 8 waves (256 threads) per block
        sage_gemm_l1_kernel<<<blocks, 256, 0, stream>>>(agg, x, Wl, bl, Wr, br, out, N);
    }
}